// GCNModel_50946902065605
// MI455X (gfx1250) — compile-verified
//
#include <hip/hip_runtime.h>
#include <hip/hip_bf16.h>

// ChebConv(K=5) x2 -> SELU -> global_add_pool -> FC for MI455X (gfx1250).
// GEMMs use V_WMMA_F32_16X16X4_F32 (full fp32 precision, wave32 layout),
// weights staged in LDS. Scatter props use f32 global atomics (L2-resident
// feature buffers: 25.6MB << 192MB L2). Pooling is atomic-free (batch sorted).

#define IN_CH 64
#define HID 64
#define OUT_CH 10
#define KCHEB 5
#define NUM_GRAPHS 64

typedef __attribute__((ext_vector_type(2))) float v2f;
typedef __attribute__((ext_vector_type(8))) float v8f;

// ---------------- elementwise helpers ----------------

__global__ void zero_f4(float4* __restrict__ p, long long n4) {
  long long i = (long long)blockIdx.x * blockDim.x + threadIdx.x;
  if (i < n4) p[i] = make_float4(0.f, 0.f, 0.f, 0.f);
}

__global__ void negcopy_f4(float4* __restrict__ dst, const float4* __restrict__ src,
                           long long n4) {
  long long i = (long long)blockIdx.x * blockDim.x + threadIdx.x;
  if (i < n4) {
    float4 v = src[i];
    dst[i] = make_float4(-v.x, -v.y, -v.z, -v.w);
  }
}

// deg[s] += 1 for every edge source s
__global__ void deg_kernel(const int* __restrict__ src, float* __restrict__ deg, int E) {
  int i = blockIdx.x * blockDim.x + threadIdx.x;
  if (i < E) atomicAdd(&deg[src[i]], 1.0f);
}

__global__ void inv_kernel(float* __restrict__ deg, int N) {
  int i = blockIdx.x * blockDim.x + threadIdx.x;
  if (i < N) {
    float d = deg[i];
    deg[i] = (d > 0.f) ? (1.0f / d) : 0.0f;
  }
}

// out[dst[e], :] += scale * (-invdeg[src[e]]) * t[src[e], :]
// 16 threads per edge, float4 per thread (64 channels). Gathers are coalesced
// (one edge row = 256B contiguous) and hit L2 (feature buffer is L2-resident).
__global__ __launch_bounds__(256) void prop_kernel(
    const float* __restrict__ t, const int* __restrict__ src,
    const int* __restrict__ dst, const float* __restrict__ invdeg,
    float* __restrict__ out, float scale, long long nwork) {
  long long gid = (long long)blockIdx.x * blockDim.x + threadIdx.x;
  if (gid >= nwork) return;
  int e = (int)(gid >> 4);
  int q = (int)(gid & 15);
  int s = src[e];
  int d = dst[e];
  float w = -scale * invdeg[s];
  float4 v = *(const float4*)(t + (long long)s * 64 + q * 4);
  float* o = out + (long long)d * 64 + q * 4;
  atomicAdd(o + 0, w * v.x);
  atomicAdd(o + 1, w * v.y);
  atomicAdd(o + 2, w * v.z);
  atomicAdd(o + 3, w * v.w);
}

// ---------------- WMMA GEMM: Y[M,64] (+)= X[M,64] @ Wg[64,64] ----------------
// 256 threads = 8 waves; wave w -> colTile (w&3), rowTile (blockIdx*2 + (w>>2)).
// M must be a multiple of 32 per block pair (100000 = 16*6250 exactly).
__global__ __launch_bounds__(256) void gemm64_wmma(
    const float* __restrict__ X, const float* __restrict__ Wg,
    float* __restrict__ Y, int beta, int M) {
  __shared__ float sW[64 * 64];
  // stage 16KB weight tile into LDS
  for (int i = threadIdx.x; i < 64 * 64 / 4; i += 256)
    ((float4*)sW)[i] = ((const float4*)Wg)[i];
  __syncthreads();

  int wave = threadIdx.x >> 5;
  int lane = threadIdx.x & 31;
  int half = lane >> 4;  // 0: K pair {k,k+1}, 1: {k+2,k+3}
  int l16  = lane & 15;
  int n0 = (wave & 3) * 16;
  int m0 = (blockIdx.x * 2 + (wave >> 2)) * 16;
  if (m0 >= M) return;  // wave-uniform; EXEC stays all-ones for WMMA

  v8f c;
#pragma unroll
  for (int j = 0; j < 8; ++j)
    c[j] = beta ? Y[(m0 + j + 8 * half) * 64 + n0 + l16] : 0.0f;

  const float* arow = X + (long long)(m0 + l16) * 64 + 2 * half;
#pragma unroll
  for (int kb = 0; kb < 64; kb += 4) {
    v2f a, b;
    a[0] = arow[kb];
    a[1] = arow[kb + 1];
    b[0] = sW[(kb + 2 * half) * 64 + n0 + l16];
    b[1] = sW[(kb + 2 * half + 1) * 64 + n0 + l16];
    c = __builtin_amdgcn_wmma_f32_16x16x4_f32(false, a, false, b, (short)0, c,
                                              false, false);
  }

#pragma unroll
  for (int j = 0; j < 8; ++j)
    Y[(m0 + j + 8 * half) * 64 + n0 + l16] = c[j];
}

// ---------------- SELU(+bias), pooling, FC ----------------

__global__ void selu_bias(float* __restrict__ y, const float* __restrict__ b,
                          long long n) {
  const float kScale = 1.0507009873554805f;
  const float kAlpha = 1.6732632423543772f;
  long long i = (long long)blockIdx.x * blockDim.x + threadIdx.x;
  if (i < n) {
    float v = y[i] + b[i & 63];
    y[i] = kScale * (v > 0.f ? v : kAlpha * (expf(v) - 1.0f));
  }
}

__device__ __forceinline__ int lower_bound(const int* a, int n, int v) {
  int lo = 0, hi = n;
  while (lo < hi) {
    int mid = (lo + hi) >> 1;
    if (a[mid] < v) lo = mid + 1; else hi = mid;
  }
  return lo;
}

// One block per graph; batch is sorted -> binary-search the node range.
// Thread t accumulates channel (t&63) over nodes strided by 4; LDS reduce.
__global__ __launch_bounds__(256) void pool_kernel(
    const float* __restrict__ h, const int* __restrict__ batch,
    float* __restrict__ g, int N) {
  int graph = blockIdx.x;
  int lo = lower_bound(batch, N, graph);
  int hi = lower_bound(batch, N, graph + 1);
  int c = threadIdx.x & 63;
  int s = threadIdx.x >> 6;  // 0..3
  float acc = 0.f;
  for (int i = lo + s; i < hi; i += 4) acc += h[(long long)i * 64 + c];
  __shared__ float red[256];
  red[threadIdx.x] = acc;
  __syncthreads();
  if (s == 0) g[graph * 64 + c] = red[c] + red[64 + c] + red[128 + c] + red[192 + c];
}

__global__ void fc_kernel(const float* __restrict__ g, const float* __restrict__ W,
                          const float* __restrict__ b, float* __restrict__ out) {
  int idx = blockIdx.x * blockDim.x + threadIdx.x;
  if (idx < NUM_GRAPHS * OUT_CH) {
    int i = idx / OUT_CH, j = idx % OUT_CH;
    float acc = b[j];
#pragma unroll 8
    for (int k = 0; k < HID; ++k) acc += g[i * 64 + k] * W[k * OUT_CH + j];
    out[idx] = acc;
  }
}

// ---------------- host orchestration ----------------

static inline long long cdiv(long long a, long long b) { return (a + b - 1) / b; }

static void cheb_layer(const float* xin, const float* W, const float* b,
                       float* out, float* R0, float* R1, float* R2,
                       const int* src, const int* dst, const float* invdeg,
                       int N, int E, hipStream_t stream) {
  const long long nf = (long long)N * 64;
  const long long n4 = nf / 4;
  const long long pw = (long long)E * 16;
  const int gemmBlocks = N / 32;  // N = 100000 -> 3125

  // k = 0
  gemm64_wmma<<<gemmBlocks, 256, 0, stream>>>(xin, W + 0 * 64 * 64, out, 0, N);
  // k = 1: Tx1 = prop(x)
  zero_f4<<<(int)cdiv(n4, 256), 256, 0, stream>>>((float4*)R1, n4);
  prop_kernel<<<(int)cdiv(pw, 256), 256, 0, stream>>>(xin, src, dst, invdeg, R1, 1.0f, pw);
  gemm64_wmma<<<gemmBlocks, 256, 0, stream>>>(R1, W + 1 * 64 * 64, out, 1, N);
  // k = 2: R0 = 2*prop(R1) - xin
  negcopy_f4<<<(int)cdiv(n4, 256), 256, 0, stream>>>((float4*)R0, (const float4*)xin, n4);
  prop_kernel<<<(int)cdiv(pw, 256), 256, 0, stream>>>(R1, src, dst, invdeg, R0, 2.0f, pw);
  gemm64_wmma<<<gemmBlocks, 256, 0, stream>>>(R0, W + 2 * 64 * 64, out, 1, N);
  // k = 3: R2 = 2*prop(R0) - R1
  negcopy_f4<<<(int)cdiv(n4, 256), 256, 0, stream>>>((float4*)R2, (const float4*)R1, n4);
  prop_kernel<<<(int)cdiv(pw, 256), 256, 0, stream>>>(R0, src, dst, invdeg, R2, 2.0f, pw);
  gemm64_wmma<<<gemmBlocks, 256, 0, stream>>>(R2, W + 3 * 64 * 64, out, 1, N);
  // k = 4: R1 = 2*prop(R2) - R0
  negcopy_f4<<<(int)cdiv(n4, 256), 256, 0, stream>>>((float4*)R1, (const float4*)R0, n4);
  prop_kernel<<<(int)cdiv(pw, 256), 256, 0, stream>>>(R2, src, dst, invdeg, R1, 2.0f, pw);
  gemm64_wmma<<<gemmBlocks, 256, 0, stream>>>(R1, W + 4 * 64 * 64, out, 1, N);

  selu_bias<<<(int)cdiv(nf, 256), 256, 0, stream>>>(out, b, nf);
}

extern "C" void kernel_launch(void* const* d_in, const int* in_sizes, int n_in,
                              void* d_out, int out_size, void* d_ws, size_t ws_size,
                              hipStream_t stream) {
  const float* x    = (const float*)d_in[0];
  const int*   ei   = (const int*)d_in[1];
  const int*   batch= (const int*)d_in[2];
  const float* W1   = (const float*)d_in[3];
  const float* b1   = (const float*)d_in[4];
  const float* W2   = (const float*)d_in[5];
  const float* b2   = (const float*)d_in[6];
  const float* Wfc  = (const float*)d_in[7];
  const float* bfc  = (const float*)d_in[8];
  float* out = (float*)d_out;

  const int N = in_sizes[0] / IN_CH;   // 100000
  const int E = in_sizes[1] / 2;       // 1250000
  const int* src = ei;
  const int* dst = ei + E;
  const long long nf = (long long)N * 64;

  // workspace carve-up (~129 MB): invdeg | R0 | R1 | R2 | H1 | O2 | G
  float* ws     = (float*)d_ws;
  float* invdeg = ws;
  float* R0     = invdeg + N;
  float* R1     = R0 + nf;
  float* R2     = R1 + nf;
  float* H1     = R2 + nf;
  float* O2     = H1 + nf;
  float* G      = O2 + nf;

  // degrees -> inverse degrees (re-init every call; ws is not preserved)
  zero_f4<<<(int)cdiv(N / 4, 256), 256, 0, stream>>>((float4*)invdeg, N / 4);
  deg_kernel<<<(int)cdiv(E, 256), 256, 0, stream>>>(src, invdeg, E);
  inv_kernel<<<(int)cdiv(N, 256), 256, 0, stream>>>(invdeg, N);

  // layer 1: H1 = selu(cheb(x; W1) + b1)
  cheb_layer(x, W1, b1, H1, R0, R1, R2, src, dst, invdeg, N, E, stream);
  // layer 2: O2 = selu(cheb(H1; W2) + b2)
  cheb_layer(H1, W2, b2, O2, R0, R1, R2, src, dst, invdeg, N, E, stream);

  // global add pool (no atomics: batch is sorted) + FC
  pool_kernel<<<NUM_GRAPHS, 256, 0, stream>>>(O2, batch, G, N);
  fc_kernel<<<(int)cdiv(NUM_GRAPHS * OUT_CH, 256), 256, 0, stream>>>(G, Wfc, bfc, out);
}